// channels_73976516706321
// MI455X (gfx1250) — compile-verified
//
#include <hip/hip_runtime.h>

typedef float v2f __attribute__((ext_vector_type(2)));
typedef float v4f __attribute__((ext_vector_type(4)));
typedef float v8f __attribute__((ext_vector_type(8)));

#define C_DIM 256
#define H_DIM 128
#define B_DIM 32
#define HW    4096   // 64*64

// ---------------------------------------------------------------------------
// Kernel 1: fused global max + avg pool over spatial dims.
// One 256-thread block per (b,c) pair; 4096 contiguous floats per pair.
// pooled layout: rows 0..31 = max, rows 32..63 = avg  -> pooled[64][256]
// ---------------------------------------------------------------------------
__global__ __launch_bounds__(256) void ca_pool(const float* __restrict__ x,
                                               float* __restrict__ pooled) {
  const int bc = blockIdx.x;                       // b*C + c
  const v4f* p4 = (const v4f*)(x + (size_t)bc * HW);

  float mx  = -3.402823466e38f;
  float sum = 0.f;
#pragma unroll
  for (int i = 0; i < 4; ++i) {
    v4f v = p4[threadIdx.x + 256 * i];             // global_load_b128
    mx  = fmaxf(mx, fmaxf(fmaxf(v.x, v.y), fmaxf(v.z, v.w)));
    sum += (v.x + v.y) + (v.z + v.w);
  }
  // wave32 butterfly reduce
#pragma unroll
  for (int off = 16; off > 0; off >>= 1) {
    mx  = fmaxf(mx, __shfl_xor(mx, off, 32));
    sum += __shfl_xor(sum, off, 32);
  }
  __shared__ float smx[8];
  __shared__ float ssm[8];
  const int wave = threadIdx.x >> 5;
  if ((threadIdx.x & 31) == 0) { smx[wave] = mx; ssm[wave] = sum; }
  __syncthreads();
  if (threadIdx.x == 0) {
    float m = smx[0], s = ssm[0];
#pragma unroll
    for (int w = 1; w < 8; ++w) { m = fmaxf(m, smx[w]); s += ssm[w]; }
    const int b = bc >> 8, c = bc & 255;
    pooled[b * C_DIM + c]           = m;                     // max rows
    pooled[(b + B_DIM) * C_DIM + c] = s * (1.f / (float)HW); // avg rows
  }
}

// ---------------------------------------------------------------------------
// Kernel 2: shared MLP on the 64 pooled rows with fp32 WMMA, then
// gate = sigmoid(mlp(mx) + mlp(av)).
// GEMM1: h[64,128] = relu(pooled[64,256] @ w1^T + b1)
// GEMM2: out[64,256] = h @ w2^T (+ 2*b2 folded into the mx/av combine)
// One block, 8 waves, uniform control flow (EXEC all-1s around WMMA).
//
// V_WMMA_F32_16X16X4_F32 operand layout (ISA 7.12.2):
//   A 16x4 : lane l (l<16: khalf=0, l>=16: khalf=1), VGPR j -> A[l%16][khalf*2+j]
//   B 4x16 : lane l, VGPR j -> B[khalf*2+j][l%16]
//   C/D    : VGPR v, lane l -> D[v + 8*(l/16)][l%16]
// ---------------------------------------------------------------------------
__global__ __launch_bounds__(256) void ca_mlp_gate(const float* __restrict__ pooled,
                                                   const float* __restrict__ w1,
                                                   const float* __restrict__ b1,
                                                   const float* __restrict__ w2,
                                                   const float* __restrict__ b2,
                                                   float* __restrict__ gate) {
  __shared__ float h[64][H_DIM + 1];               // +1 pad: kill 64-bank conflicts
  const int lane = threadIdx.x & 31;
  const int wave = threadIdx.x >> 5;
  const int ln   = lane & 15;                      // M or N position in tile
  const int lh   = lane >> 4;                      // K half select

  // ---- GEMM1: 4 row-tiles x 8 col-tiles = 32 tiles, 4 per wave ----
#pragma unroll
  for (int i = 0; i < 4; ++i) {
    const int t  = wave + i * 8;                   // 0..31, uniform trip count
    const int mt = t >> 3;                         // 0..3
    const int nt = t & 7;                          // 0..7
    const int m  = mt * 16 + ln;                   // pooled row
    const int n  = nt * 16 + ln;                   // hidden unit
    v8f acc = {};
    for (int k0 = 0; k0 < C_DIM; k0 += 4) {
      const int ka = k0 + lh * 2;
      v2f a, bf;
      a.x  = pooled[m * C_DIM + ka];
      a.y  = pooled[m * C_DIM + ka + 1];
      bf.x = w1[n * C_DIM + ka];                   // w1^T[k][n] = w1[n][k]
      bf.y = w1[n * C_DIM + ka + 1];
      acc = __builtin_amdgcn_wmma_f32_16x16x4_f32(
          false, a, false, bf, (short)0, acc, false, false);
    }
    const float bias = b1[n];
#pragma unroll
    for (int v = 0; v < 8; ++v)
      h[mt * 16 + v + 8 * lh][n] = fmaxf(acc[v] + bias, 0.f);
  }
  __syncthreads();

  // ---- GEMM2: pair mx row-tile (rt) with av row-tile (rt+2), combine in-reg.
  // 2 row-tiles x 16 col-tiles = 32 pairs, 4 per wave.
#pragma unroll
  for (int i = 0; i < 4; ++i) {
    const int t  = wave + i * 8;                   // 0..31
    const int rt = t >> 4;                         // 0..1 (mx half)
    const int nt = t & 15;                         // 0..15
    const int n  = nt * 16 + ln;                   // output channel
    const int mm = rt * 16 + ln;                   // mx row (0..31)
    v8f accm = {};
    v8f acca = {};
    for (int k0 = 0; k0 < H_DIM; k0 += 4) {
      const int ka = k0 + lh * 2;
      v2f am, aa, bf;
      am.x = h[mm][ka];        am.y = h[mm][ka + 1];
      aa.x = h[mm + 32][ka];   aa.y = h[mm + 32][ka + 1];
      bf.x = w2[n * H_DIM + ka];                   // w2^T[k][n] = w2[n][k]
      bf.y = w2[n * H_DIM + ka + 1];
      accm = __builtin_amdgcn_wmma_f32_16x16x4_f32(
          false, am, false, bf, (short)0, accm, false, false);
      acca = __builtin_amdgcn_wmma_f32_16x16x4_f32(
          false, aa, false, bf, (short)0, acca, false, false);
    }
    const float bias2 = 2.f * b2[n];               // b2 appears in both MLPs
#pragma unroll
    for (int v = 0; v < 8; ++v) {
      const int brow = rt * 16 + v + 8 * lh;       // batch index 0..31
      const float s  = accm[v] + acca[v] + bias2;
      gate[brow * C_DIM + n] = 1.f / (1.f + __expf(-s));
    }
  }
}

// ---------------------------------------------------------------------------
// Kernel 3: out = gate[b,c] * x. Streaming: NT load of x (last use, it's in
// L2 from kernel 1), NT store of out (don't pollute L2).
// ---------------------------------------------------------------------------
__global__ __launch_bounds__(256) void ca_scale(const float* __restrict__ x,
                                                const float* __restrict__ gate,
                                                float* __restrict__ out) {
  const int bc = blockIdx.x;
  const float g = gate[bc];                        // uniform per block
  const v4f* xin = (const v4f*)(x + (size_t)bc * HW);
  v4f* o = (v4f*)(out + (size_t)bc * HW);
#pragma unroll
  for (int i = 0; i < 4; ++i) {
    const int idx = threadIdx.x + 256 * i;
    v4f v = __builtin_nontemporal_load(&xin[idx]);
    v4f r = v * g;
    __builtin_nontemporal_store(r, &o[idx]);
  }
}

// ---------------------------------------------------------------------------
extern "C" void kernel_launch(void* const* d_in, const int* in_sizes, int n_in,
                              void* d_out, int out_size, void* d_ws, size_t ws_size,
                              hipStream_t stream) {
  const float* x  = (const float*)d_in[0];   // [32,256,64,64]
  const float* w1 = (const float*)d_in[1];   // [128,256]
  const float* b1 = (const float*)d_in[2];   // [128]
  const float* w2 = (const float*)d_in[3];   // [256,128]
  const float* b2 = (const float*)d_in[4];   // [256]
  float* out = (float*)d_out;

  float* pooled = (float*)d_ws;              // 64*256 floats
  float* gate   = pooled + 64 * C_DIM;       // 32*256 floats

  ca_pool<<<B_DIM * C_DIM, 256, 0, stream>>>(x, pooled);
  ca_mlp_gate<<<1, 256, 0, stream>>>(pooled, w1, b1, w2, b2, gate);
  ca_scale<<<B_DIM * C_DIM, 256, 0, stream>>>(x, gate, out);
}